// GINEEncoder_3375844295314
// MI455X (gfx1250) — compile-verified
//
#include <hip/hip_runtime.h>
#include <hip/hip_bf16.h>

// ---------------------------------------------------------------------------
// GINE forward for MI455X (gfx1250, wave32).
// GEMMs on the matrix pipe via V_WMMA_F32_16X16X4_F32 (fp32-exact).
// Weights staged into LDS pair-interleaved via GLOBAL_LOAD_ASYNC_TO_LDS_B32
// (ASYNCcnt) so every B fragment is one ds_load_b64.
// Edge stage fuses edge-GEMM + gather + relu + scatter (global_atomic_add_f32,
// L2-resident since h/agg are 25.6 MB each vs 192 MB L2).
// BatchNorm = fused per-column atomic {sum,sumsq} + tiny finalize kernels.
// ---------------------------------------------------------------------------

typedef __attribute__((ext_vector_type(2))) float v2f;
typedef __attribute__((ext_vector_type(8))) float v8f;

#define N_NODES    100000
#define N_EDGES    1600000
#define NUM_FEAT   128
#define EDGE_FEAT  32
#define HID        64
#define HID2       128
#define LAYERS     3
#define NUM_GRAPHS 512
#define BN_EPS     1e-5f

static __device__ __forceinline__ v8f v8f_zero() {
  v8f v = {0.f,0.f,0.f,0.f,0.f,0.f,0.f,0.f};
  return v;
}

#if defined(__has_builtin)
#if __has_builtin(__builtin_amdgcn_wmma_f32_16x16x4_f32)
#define HAVE_WMMA_F32K4 1
#endif
#endif

// D = A(16x4,f32) * B(4x16,f32) + C(16x16,f32), wave32.
// A frag: lane(l16,half) holds A[m=l16][k0+2*half + {0,1}]  (contiguous pair)
// B frag: lane(l16,half) holds B[k0+2*half + {0,1}][n=l16]
// C/D   : vgpr r, lane(l16,half) = C[row = r + 8*half][col = l16]
static __device__ __forceinline__ v8f wmma_k4(v2f a, v2f b, v8f c) {
#ifdef HAVE_WMMA_F32K4
  return __builtin_amdgcn_wmma_f32_16x16x4_f32(
      /*neg_a=*/false, a, /*neg_b=*/false, b,
      /*c_mod=*/(short)0, c, /*reuse_a=*/false, /*reuse_b=*/false);
#else
  c[0] += a[0] * b[0] + a[1] * b[1];
  return c;
#endif
}

// ---------------------------------------------------------------------------
// Stage a K x Nw row-major weight matrix into LDS, pair-interleaved over k:
//   lds[((k>>1)*Nw + n)*2 + (k&1)] = W[k*Nw + n]
// so a B fragment (rows kk,kk+1 at column n) is ONE aligned ds_load_b64.
// Uses GLOBAL_LOAD_ASYNC_TO_LDS_B32: per-lane global->LDS DMA, ASYNCcnt.
static __device__ __forceinline__ void stage_pairs_async(float* lds,
                                                         const float* g,
                                                         int total, int logNw) {
  const int nmask = (1 << logNw) - 1;
  for (int i = threadIdx.x; i < total; i += 256) {
    const int k = i >> logNw;
    const int n = i & nmask;
    const int di = ((((k >> 1) << logNw) + n) << 1) + (k & 1);
    const unsigned lofs = (unsigned)(size_t)(lds + di);  // LDS byte address
    const float* gp = g + i;
    asm volatile("global_load_async_to_lds_b32 %0, %1, off"
                 :: "v"(lofs), "v"(gp) : "memory");
  }
  // each wave drains its own async ops before signaling the barrier
  asm volatile("s_wait_asynccnt 0x0" ::: "memory");
}

// B fragment fetch from pair-interleaved LDS: one ds_load_b64
static __device__ __forceinline__ v2f lds_bfrag(const float* lds, int pairIdx,
                                                int col, int logNw) {
  return *(const v2f*)(lds + ((((pairIdx << logNw) + col)) << 1));
}

// ---------------------------------------------------------------------------
__global__ __launch_bounds__(256) void zero_kernel(float* __restrict__ p, long n) {
  long i = (long)blockIdx.x * blockDim.x + threadIdx.x;
  if (i < n) p[i] = 0.f;
}

// ---------------------------------------------------------------------------
// h = x @ enc_W + enc_b      [N,128] @ [128,64]
__global__ __launch_bounds__(256)
void encoder_kernel(const float* __restrict__ x, const float* __restrict__ W,
                    const float* __restrict__ b, float* __restrict__ h, int n) {
  __shared__ float Wlds[NUM_FEAT * HID];   // 32 KB, pair-interleaved
  __shared__ float blds[HID];
  stage_pairs_async(Wlds, W, NUM_FEAT * HID, /*logNw=*/6);
  for (int i = threadIdx.x; i < HID; i += 256) blds[i] = b[i];
  __syncthreads();

  const int wave = threadIdx.x >> 5, lane = threadIdx.x & 31;
  const int half = lane >> 4, l16 = lane & 15;
  const int row0 = (blockIdx.x * 8 + wave) * 16;
  const int m = row0 + l16;
  // clamp instead of predicate: out-of-range rows compute garbage that is
  // never stored (keeps A loads unconditional global_load_b64)
  const float* arow = x + (size_t)(m < n ? m : 0) * NUM_FEAT + 2 * half;

  v8f acc[4];
#pragma unroll
  for (int t = 0; t < 4; ++t) acc[t] = v8f_zero();

#pragma unroll 4
  for (int k0 = 0; k0 < NUM_FEAT; k0 += 4) {
    const v2f a = *(const v2f*)(arow + k0);
    const int pairIdx = (k0 >> 1) + half;
#pragma unroll
    for (int t = 0; t < 4; ++t)
      acc[t] = wmma_k4(a, lds_bfrag(Wlds, pairIdx, t * 16 + l16, 6), acc[t]);
  }

#pragma unroll
  for (int r = 0; r < 8; ++r) {
    const int rr = row0 + r + 8 * half;
    if (rr < n) {
#pragma unroll
      for (int t = 0; t < 4; ++t) {
        const int col = t * 16 + l16;
        h[(size_t)rr * HID + col] = acc[t][r] + blds[col];
      }
    }
  }
}

// ---------------------------------------------------------------------------
// fused edge stage: e = edge_attr @ Wl + bl ; msg = relu(h[src]+e) ;
// agg[dst] += msg.  one wave = 16 edges; block = 128 edges; E % 128 == 0
__global__ __launch_bounds__(256)
void edge_kernel(const float* __restrict__ eattr, const int* __restrict__ src,
                 const int* __restrict__ dst, const float* __restrict__ Wl,
                 const float* __restrict__ bl, const float* __restrict__ h,
                 float* __restrict__ agg) {
  __shared__ float Wlds[EDGE_FEAT * HID];  // 8 KB, pair-interleaved
  __shared__ float blds[HID];
  stage_pairs_async(Wlds, Wl, EDGE_FEAT * HID, /*logNw=*/6);
  for (int i = threadIdx.x; i < HID; i += 256) blds[i] = bl[i];
  __syncthreads();

  const int wave = threadIdx.x >> 5, lane = threadIdx.x & 31;
  const int half = lane >> 4, l16 = lane & 15;
  const long e0 = ((long)blockIdx.x * 8 + wave) * 16;

  const float* arow = eattr + (e0 + l16) * EDGE_FEAT + 2 * half;
  __builtin_prefetch(arow + 128 * EDGE_FEAT, 0, 0);  // global_prefetch_b8

  v8f acc[4];
#pragma unroll
  for (int t = 0; t < 4; ++t) acc[t] = v8f_zero();

#pragma unroll
  for (int k0 = 0; k0 < EDGE_FEAT; k0 += 4) {
    const v2f a = *(const v2f*)(arow + k0);
    const int pairIdx = (k0 >> 1) + half;
#pragma unroll
    for (int t = 0; t < 4; ++t)
      acc[t] = wmma_k4(a, lds_bfrag(Wlds, pairIdx, t * 16 + l16, 6), acc[t]);
  }

#pragma unroll
  for (int r = 0; r < 8; ++r) {
    const long e = e0 + r + 8 * half;
    const int s = src[e];
    const int d = dst[e];
    const float* hrow = h + (size_t)s * HID;
    float* grow = agg + (size_t)d * HID;
#pragma unroll
    for (int t = 0; t < 4; ++t) {
      const int col = t * 16 + l16;
      float v = acc[t][r] + blds[col] + hrow[col];
      v = fmaxf(v, 0.f);
      atomicAdd(grow + col, v);
    }
  }
}

// ---------------------------------------------------------------------------
// t1 = ((1+eps)*h + agg) @ W1 + b1     [N,64] @ [64,128]  + BN1 stats
__global__ __launch_bounds__(256)
void mlp1_kernel(const float* __restrict__ h, const float* __restrict__ agg,
                 const float* __restrict__ epsArr, int layer,
                 const float* __restrict__ W1, const float* __restrict__ b1,
                 float* __restrict__ t1, float* __restrict__ sum1,
                 float* __restrict__ sq1, int n) {
  __shared__ float Wlds[HID * HID2];  // 32 KB, pair-interleaved, Nw=128
  __shared__ float blds[HID2];
  stage_pairs_async(Wlds, W1, HID * HID2, /*logNw=*/7);
  for (int i = threadIdx.x; i < HID2; i += 256) blds[i] = b1[i];
  __syncthreads();

  const float ep = 1.0f + epsArr[layer];
  const int wave = threadIdx.x >> 5, lane = threadIdx.x & 31;
  const int half = lane >> 4, l16 = lane & 15;
  const int row0 = (blockIdx.x * 8 + wave) * 16;
  const int m = row0 + l16;
  const size_t abase = (size_t)(m < n ? m : 0) * HID + 2 * half;

  v8f acc[8];
#pragma unroll
  for (int t = 0; t < 8; ++t) acc[t] = v8f_zero();

#pragma unroll 4
  for (int k0 = 0; k0 < HID; k0 += 4) {
    const size_t i0 = abase + k0;
    const v2f hv = *(const v2f*)(h + i0);
    const v2f gv = *(const v2f*)(agg + i0);
    v2f a;
    a[0] = ep * hv[0] + gv[0];
    a[1] = ep * hv[1] + gv[1];
    const int pairIdx = (k0 >> 1) + half;
#pragma unroll
    for (int t = 0; t < 8; ++t)
      acc[t] = wmma_k4(a, lds_bfrag(Wlds, pairIdx, t * 16 + l16, 7), acc[t]);
  }

#pragma unroll
  for (int t = 0; t < 8; ++t) {
    const int col = t * 16 + l16;
    float s = 0.f, q = 0.f;
#pragma unroll
    for (int r = 0; r < 8; ++r) {
      const int rr = row0 + r + 8 * half;
      if (rr < n) {
        const float v = acc[t][r] + blds[col];
        t1[(size_t)rr * HID2 + col] = v;
        s += v;
        q += v * v;
      }
    }
    atomicAdd(sum1 + col, s);
    atomicAdd(sq1 + col, q);
  }
}

// ---------------------------------------------------------------------------
__global__ void bn_finalize_kernel(const float* __restrict__ sum,
                                   const float* __restrict__ sq,
                                   const float* __restrict__ g,
                                   const float* __restrict__ b,
                                   float* __restrict__ scale,
                                   float* __restrict__ shift, int C, float invN) {
  const int c = blockIdx.x * blockDim.x + threadIdx.x;
  if (c < C) {
    const float m = sum[c] * invN;
    const float v = sq[c] * invN - m * m;
    const float s = g[c] * rsqrtf(v + BN_EPS);
    scale[c] = s;
    shift[c] = b[c] - s * m;
  }
}

// ---------------------------------------------------------------------------
// t2 = relu(bn1(t1)) @ W2 + b2      [N,128] @ [128,64]  + BN2 stats
__global__ __launch_bounds__(256)
void mlp2_kernel(const float* __restrict__ t1, const float* __restrict__ scale1,
                 const float* __restrict__ shift1, const float* __restrict__ W2,
                 const float* __restrict__ b2, float* __restrict__ t2,
                 float* __restrict__ sum2, float* __restrict__ sq2, int n) {
  __shared__ float Wlds[HID2 * HID];  // 32 KB, pair-interleaved, Nw=64
  __shared__ float blds[HID];
  __shared__ float sc[HID2];
  __shared__ float sh[HID2];
  stage_pairs_async(Wlds, W2, HID2 * HID, /*logNw=*/6);
  for (int i = threadIdx.x; i < HID; i += 256) blds[i] = b2[i];
  for (int i = threadIdx.x; i < HID2; i += 256) { sc[i] = scale1[i]; sh[i] = shift1[i]; }
  __syncthreads();

  const int wave = threadIdx.x >> 5, lane = threadIdx.x & 31;
  const int half = lane >> 4, l16 = lane & 15;
  const int row0 = (blockIdx.x * 8 + wave) * 16;
  const int m = row0 + l16;
  const size_t abase = (size_t)(m < n ? m : 0) * HID2 + 2 * half;

  v8f acc[4];
#pragma unroll
  for (int t = 0; t < 4; ++t) acc[t] = v8f_zero();

#pragma unroll 4
  for (int k0 = 0; k0 < HID2; k0 += 4) {
    const int kk = k0 + 2 * half;
    const v2f tv = *(const v2f*)(t1 + abase + k0);
    v2f a;
    a[0] = fmaxf(sc[kk] * tv[0] + sh[kk], 0.f);
    a[1] = fmaxf(sc[kk + 1] * tv[1] + sh[kk + 1], 0.f);
    const int pairIdx = (k0 >> 1) + half;
#pragma unroll
    for (int t = 0; t < 4; ++t)
      acc[t] = wmma_k4(a, lds_bfrag(Wlds, pairIdx, t * 16 + l16, 6), acc[t]);
  }

#pragma unroll
  for (int t = 0; t < 4; ++t) {
    const int col = t * 16 + l16;
    float s = 0.f, q = 0.f;
#pragma unroll
    for (int r = 0; r < 8; ++r) {
      const int rr = row0 + r + 8 * half;
      if (rr < n) {
        const float v = acc[t][r] + blds[col];
        t2[(size_t)rr * HID + col] = v;
        s += v;
        q += v * v;
      }
    }
    atomicAdd(sum2 + col, s);
    atomicAdd(sq2 + col, q);
  }
}

// ---------------------------------------------------------------------------
__global__ __launch_bounds__(256)
void bn_relu_kernel(const float* __restrict__ t2, const float* __restrict__ scale2,
                    const float* __restrict__ shift2, float* __restrict__ h, long total) {
  const long i = (long)blockIdx.x * blockDim.x + threadIdx.x;
  if (i < total) {
    const int c = (int)(i & (HID - 1));
    h[i] = fmaxf(scale2[c] * t2[i] + shift2[c], 0.f);
  }
}

// ---------------------------------------------------------------------------
__global__ __launch_bounds__(256)
void pool_acc_kernel(const float* __restrict__ h, const int* __restrict__ batch,
                     float* __restrict__ pool, int n) {
  const long i = (long)blockIdx.x * blockDim.x + threadIdx.x;
  if (i < (long)n * HID) {
    const int nd = (int)(i >> 6);
    const int c = (int)(i & (HID - 1));
    atomicAdd(pool + (size_t)batch[nd] * HID + c, h[i]);
  }
}

__global__ __launch_bounds__(256)
void pool_cnt_kernel(const int* __restrict__ batch, float* __restrict__ cnt, int n) {
  const int i = blockIdx.x * blockDim.x + threadIdx.x;
  if (i < n) atomicAdd(cnt + batch[i], 1.f);
}

__global__ __launch_bounds__(256)
void pool_div_kernel(const float* __restrict__ pool, const float* __restrict__ cnt,
                     float* __restrict__ out) {
  const int i = blockIdx.x * blockDim.x + threadIdx.x;
  if (i < NUM_GRAPHS * HID) {
    const int g = i >> 6;
    out[i] = pool[i] / fmaxf(cnt[g], 1.f);
  }
}

// ---------------------------------------------------------------------------
extern "C" void kernel_launch(void* const* d_in, const int* in_sizes, int n_in,
                              void* d_out, int out_size, void* d_ws, size_t ws_size,
                              hipStream_t stream) {
  const float* x      = (const float*)d_in[0];
  const int*   ei     = (const int*)d_in[1];
  const float* eattr  = (const float*)d_in[2];
  const int*   batch  = (const int*)d_in[3];
  const float* encW   = (const float*)d_in[4];
  const float* encb   = (const float*)d_in[5];
  const float* eps    = (const float*)d_in[6];
  const float* edgeW  = (const float*)d_in[7];
  const float* edgeb  = (const float*)d_in[8];
  const float* W1     = (const float*)d_in[9];
  const float* b1     = (const float*)d_in[10];
  const float* g1     = (const float*)d_in[11];
  const float* bb1    = (const float*)d_in[12];
  const float* W2     = (const float*)d_in[13];
  const float* b2     = (const float*)d_in[14];
  const float* g2     = (const float*)d_in[15];
  const float* bb2    = (const float*)d_in[16];

  const int* src = ei;
  const int* dst = ei + N_EDGES;

  float* ws = (float*)d_ws;
  float* h      = ws; ws += (size_t)N_NODES * HID;
  float* agg    = ws; ws += (size_t)N_NODES * HID;   // reused as t2
  float* t1     = ws; ws += (size_t)N_NODES * HID2;
  float* sum1   = ws; ws += HID2;                    // sum1,sq1,sum2,sq2 contiguous
  float* sq1    = ws; ws += HID2;
  float* sum2   = ws; ws += HID;
  float* sq2    = ws; ws += HID;
  float* scale1 = ws; ws += HID2;
  float* shift1 = ws; ws += HID2;
  float* scale2 = ws; ws += HID;
  float* shift2 = ws; ws += HID;
  float* pool   = ws; ws += (size_t)NUM_GRAPHS * HID;  // pool+cnt contiguous
  float* cnt    = ws; ws += NUM_GRAPHS;

  const int nodeBlocks = (N_NODES + 127) / 128;
  const int edgeBlocks = N_EDGES / 128;
  const long nh = (long)N_NODES * HID;

  encoder_kernel<<<nodeBlocks, 256, 0, stream>>>(x, encW, encb, h, N_NODES);

  for (int l = 0; l < LAYERS; ++l) {
    zero_kernel<<<(int)((nh + 255) / 256), 256, 0, stream>>>(agg, nh);
    zero_kernel<<<2, 256, 0, stream>>>(sum1, 2 * HID2 + 2 * HID);

    edge_kernel<<<edgeBlocks, 256, 0, stream>>>(
        eattr, src, dst, edgeW + (size_t)l * EDGE_FEAT * HID, edgeb + (size_t)l * HID,
        h, agg);

    mlp1_kernel<<<nodeBlocks, 256, 0, stream>>>(
        h, agg, eps, l, W1 + (size_t)l * HID * HID2, b1 + (size_t)l * HID2,
        t1, sum1, sq1, N_NODES);

    bn_finalize_kernel<<<1, HID2, 0, stream>>>(
        sum1, sq1, g1 + (size_t)l * HID2, bb1 + (size_t)l * HID2,
        scale1, shift1, HID2, 1.0f / N_NODES);

    mlp2_kernel<<<nodeBlocks, 256, 0, stream>>>(
        t1, scale1, shift1, W2 + (size_t)l * HID2 * HID, b2 + (size_t)l * HID,
        /*t2=*/agg, sum2, sq2, N_NODES);

    bn_finalize_kernel<<<1, HID, 0, stream>>>(
        sum2, sq2, g2 + (size_t)l * HID, bb2 + (size_t)l * HID,
        scale2, shift2, HID, 1.0f / N_NODES);

    bn_relu_kernel<<<(int)((nh + 255) / 256), 256, 0, stream>>>(
        /*t2=*/agg, scale2, shift2, h, nh);
  }

  zero_kernel<<<(NUM_GRAPHS * HID + NUM_GRAPHS + 255) / 256, 256, 0, stream>>>(
      pool, NUM_GRAPHS * HID + NUM_GRAPHS);
  pool_acc_kernel<<<(int)((nh + 255) / 256), 256, 0, stream>>>(h, batch, pool, N_NODES);
  pool_cnt_kernel<<<(N_NODES + 255) / 256, 256, 0, stream>>>(batch, cnt, N_NODES);
  pool_div_kernel<<<(NUM_GRAPHS * HID + 255) / 256, 256, 0, stream>>>(
      pool, cnt, (float*)d_out);
}